// GCNLayer_13151189860789
// MI455X (gfx1250) — compile-verified
//
#include <hip/hip_runtime.h>
#include <math.h>

typedef __attribute__((ext_vector_type(2))) float v2f;
typedef __attribute__((ext_vector_type(8))) float v8f;

#define LDW   132   // 128-wide LDS row pitch (4-bank stride, 16B aligned)
#define SLDW  34    // scores row pitch (even -> b64-aligned pairs)
#define VLDW  260   // 256-wide LDS row pitch

__device__ __forceinline__ v8f wmma_f32(v2f a, v2f b, v8f c) {
  // D(16x16,f32) = A(16x4,f32) * B(4x16,f32) + C
  return __builtin_amdgcn_wmma_f32_16x16x4_f32(false, a, false, b, (short)0, c,
                                               false, false);
}

// ---------------------------------------------------------------------------
// Generic 128-in GEMM:  out[m][o] = epi( sum_h X[m][h]*W[o][h] + bias[o] )
// Optional: row gather (e_nb fusion), residual+relu epilogue + BN sum atomics.
// Block = 256 thr = 8 waves; wave w owns output cols [16w,16w+16); grid-stride
// over 16-row M tiles. W (128x128) cached in LDS once per block.
// ---------------------------------------------------------------------------
__global__ __launch_bounds__(256)
void gemm128(const float* __restrict__ X, const float* __restrict__ W,
             const float* __restrict__ bias, float* __restrict__ out,
             int Mtiles, const int* __restrict__ gatherIdx,
             const float* __restrict__ resid,
             float* __restrict__ bnSum, float* __restrict__ bnSq) {
  __shared__ float wl[128 * LDW];
  __shared__ float xs[16 * LDW];
  const int tid = threadIdx.x;
  for (int idx = tid; idx < 128 * 32; idx += 256) {
    int row = idx >> 5, c4 = (idx & 31) << 2;
    *(float4*)&wl[row * LDW + c4] = *(const float4*)(W + (row << 7) + c4);
  }
  const int wave = tid >> 5, lane = tid & 31;
  const int lm = lane & 15, half = lane >> 4;
  const int kh = half << 1, o0 = wave << 4;
  const float bo = bias[o0 + lm];
  float lsum = 0.f, lsq = 0.f;
  __syncthreads();
  for (int mt = blockIdx.x; mt < Mtiles; mt += gridDim.x) {
    __syncthreads();
    for (int idx = tid; idx < 512; idx += 256) {   // stage 16x128 X tile
      int row = idx >> 5, c4 = (idx & 31) << 2;
      int m = (mt << 4) + row;
      long srcRow = m;
      if (gatherIdx) {                              // m = ((b*192)+j)*24+k
        int b = m / (192 * 24);
        int rem = m - b * (192 * 24);
        int j = rem / 24;
        srcRow = (long)(b * 192 + j) * 192 + gatherIdx[m];
      }
      *(float4*)&xs[row * LDW + c4] = *(const float4*)(X + (srcRow << 7) + c4);
    }
    __syncthreads();
    v8f acc = {0.f, 0.f, 0.f, 0.f, 0.f, 0.f, 0.f, 0.f};
#pragma unroll
    for (int k0 = 0; k0 < 128; k0 += 4) {
      v2f a = *(const v2f*)&xs[lm * LDW + k0 + kh];          // A: M=lm, K pair
      v2f b = *(const v2f*)&wl[(o0 + lm) * LDW + k0 + kh];   // B: N=lm, K pair
      acc = wmma_f32(a, b, acc);
    }
#pragma unroll
    for (int r = 0; r < 8; ++r) {                  // D: row r+8*half, col lm
      int m = (mt << 4) + r + (half << 3);
      long o = ((long)m << 7) + o0 + lm;
      float v = acc[r] + bo;
      if (resid) {
        v = fmaxf(v, 0.f) + resid[o];
        lsum += v; lsq += v * v;
      }
      out[o] = v;
    }
  }
  if (bnSum) {
    atomicAdd(&bnSum[o0 + lm], lsum);
    atomicAdd(&bnSq[o0 + lm], lsq);
  }
}

// ---------------------------------------------------------------------------
// Attention per (b,j): S = q_j * kk^T / sqrt(H); softmax over 24; att = A*vv.
// att overwrites q in place (block touches only its own column j).
// ---------------------------------------------------------------------------
__global__ __launch_bounds__(256)
void attn(float* __restrict__ q, const float* __restrict__ kk,
          const float* __restrict__ vv) {
  __shared__ float qs[192 * LDW];   // q column j, 192x128
  __shared__ float ks[32 * LDW];    // kk (rows 24..31 zero)
  __shared__ float vs[24 * LDW];
  __shared__ float ss[192 * SLDW];  // scores / softmax weights
  const int tid = threadIdx.x;
  const int bj = blockIdx.x;
  const int b = bj / 192, j = bj - b * 192;
  for (int idx = tid; idx < 192 * 32; idx += 256) {
    int row = idx >> 5, c4 = (idx & 31) << 2;
    long src = ((long)(b * 192 + row) * 192 + j) << 7;
    *(float4*)&qs[row * LDW + c4] = *(const float4*)(q + src + c4);
  }
  for (int idx = tid; idx < 32 * 32; idx += 256) {
    int row = idx >> 5, c4 = (idx & 31) << 2;
    float4 v = {0.f, 0.f, 0.f, 0.f};
    if (row < 24) v = *(const float4*)(kk + (((long)bj * 24 + row) << 7) + c4);
    *(float4*)&ks[row * LDW + c4] = v;
  }
  for (int idx = tid; idx < 24 * 32; idx += 256) {
    int row = idx >> 5, c4 = (idx & 31) << 2;
    *(float4*)&vs[row * LDW + c4] =
        *(const float4*)(vv + (((long)bj * 24 + row) << 7) + c4);
  }
  __syncthreads();
  const int wave = tid >> 5, lane = tid & 31;
  const int lm = lane & 15, half = lane >> 4, kh = half << 1;
  // --- scores: 12 M-tiles x 2 N-tiles = 24 wave-tiles
#pragma unroll
  for (int t = 0; t < 3; ++t) {
    int tile = wave + (t << 3);
    int it = tile >> 1, n0 = (tile & 1) << 4;
    v8f acc = {0.f, 0.f, 0.f, 0.f, 0.f, 0.f, 0.f, 0.f};
#pragma unroll
    for (int k0 = 0; k0 < 128; k0 += 4) {
      v2f a = *(const v2f*)&qs[((it << 4) + lm) * LDW + k0 + kh];
      v2f bb = *(const v2f*)&ks[(n0 + lm) * LDW + k0 + kh];
      acc = wmma_f32(a, bb, acc);
    }
#pragma unroll
    for (int r = 0; r < 8; ++r)
      ss[((it << 4) + r + (half << 3)) * SLDW + n0 + lm] =
          acc[r] * 0.08838834764831845f;  // 1/sqrt(128)
  }
  __syncthreads();
  if (tid < 192) {                      // softmax over K=24 per row
    float mx = -1e30f;
    for (int k = 0; k < 24; ++k) mx = fmaxf(mx, ss[tid * SLDW + k]);
    float sum = 0.f;
    for (int k = 0; k < 24; ++k) {
      float v = __expf(ss[tid * SLDW + k] - mx);
      ss[tid * SLDW + k] = v;
      sum += v;
    }
    float inv = 1.f / sum;
    for (int k = 0; k < 24; ++k) ss[tid * SLDW + k] *= inv;
  }
  __syncthreads();
  // --- att = A(192x24) * vv(24x128): 12 x 8 = 96 wave-tiles
#pragma unroll
  for (int t = 0; t < 12; ++t) {
    int tile = wave + (t << 3);
    int it = tile >> 3, o0 = (tile & 7) << 4;
    v8f acc = {0.f, 0.f, 0.f, 0.f, 0.f, 0.f, 0.f, 0.f};
#pragma unroll
    for (int k0 = 0; k0 < 24; k0 += 4) {
      v2f a = *(const v2f*)&ss[((it << 4) + lm) * SLDW + k0 + kh];
      v2f bb;
      bb.x = vs[(k0 + kh) * LDW + o0 + lm];
      bb.y = vs[(k0 + kh + 1) * LDW + o0 + lm];
      acc = wmma_f32(a, bb, acc);
    }
#pragma unroll
    for (int r = 0; r < 8; ++r) {
      int i = (it << 4) + r + (half << 3);
      long dst = (((long)(b * 192 + i) * 192 + j) << 7) + o0 + lm;
      q[dst] = acc[r];
    }
  }
}

// ---------------------------------------------------------------------------
// Fused gate: p = e*Vin^T+bvin; g = [p | bn1(h1)](16x256) * V^T + bv;
// out = bn1(h1) + relu(g); accumulate BN2 sums. h2 -> d_out.
// ---------------------------------------------------------------------------
__global__ __launch_bounds__(256)
void gate_fused(const float* __restrict__ e, const float* __restrict__ h1,
                const float* __restrict__ scale1, const float* __restrict__ shift1,
                const float* __restrict__ vin_w, const float* __restrict__ vin_b,
                const float* __restrict__ v_w, const float* __restrict__ v_b,
                float* __restrict__ out, int Mtiles,
                float* __restrict__ bnSum, float* __restrict__ bnSq) {
  __shared__ float vl[128 * VLDW];  // V_w 128x256
  __shared__ float xt[16 * VLDW];   // [p | h1bn] 16x256
  __shared__ float et[16 * LDW];    // e tile
  const int tid = threadIdx.x;
  for (int idx = tid; idx < 128 * 64; idx += 256) {
    int row = idx >> 6, c4 = (idx & 63) << 2;
    *(float4*)&vl[row * VLDW + c4] = *(const float4*)(v_w + (row << 8) + c4);
  }
  const int wave = tid >> 5, lane = tid & 31;
  const int lm = lane & 15, half = lane >> 4;
  const int kh = half << 1, o0 = wave << 4;
  const float bvin = vin_b[o0 + lm];
  const float bv = v_b[o0 + lm];
  float lsum = 0.f, lsq = 0.f;
  __syncthreads();
  for (int mt = blockIdx.x; mt < Mtiles; mt += gridDim.x) {
    __syncthreads();
    for (int idx = tid; idx < 512; idx += 256) {
      int row = idx >> 5, c4 = (idx & 31) << 2;
      long m = (mt << 4) + row;
      *(float4*)&et[row * LDW + c4] = *(const float4*)(e + (m << 7) + c4);
      float4 h = *(const float4*)(h1 + (m << 7) + c4);
      float4 sc = *(const float4*)(scale1 + c4);
      float4 sh = *(const float4*)(shift1 + c4);
      float4 hb;
      hb.x = h.x * sc.x + sh.x; hb.y = h.y * sc.y + sh.y;
      hb.z = h.z * sc.z + sh.z; hb.w = h.w * sc.w + sh.w;
      *(float4*)&xt[row * VLDW + 128 + c4] = hb;
    }
    __syncthreads();
    // stage 1: p tile (B operand streamed from global Vin, L2-resident)
    v8f accp = {0.f, 0.f, 0.f, 0.f, 0.f, 0.f, 0.f, 0.f};
#pragma unroll
    for (int k0 = 0; k0 < 128; k0 += 4) {
      v2f a = *(const v2f*)&et[lm * LDW + k0 + kh];
      v2f bb = *(const v2f*)(vin_w + ((long)(o0 + lm) << 7) + k0 + kh);
      accp = wmma_f32(a, bb, accp);
    }
#pragma unroll
    for (int r = 0; r < 8; ++r)
      xt[(r + (half << 3)) * VLDW + o0 + lm] = accp[r] + bvin;
    __syncthreads();
    // stage 2: K=256 GEMM over [p | h1bn]
    v8f acc = {0.f, 0.f, 0.f, 0.f, 0.f, 0.f, 0.f, 0.f};
#pragma unroll
    for (int k0 = 0; k0 < 256; k0 += 4) {
      v2f a = *(const v2f*)&xt[lm * VLDW + k0 + kh];
      v2f bb = *(const v2f*)&vl[(o0 + lm) * VLDW + k0 + kh];
      acc = wmma_f32(a, bb, acc);
    }
#pragma unroll
    for (int r = 0; r < 8; ++r) {
      int rr = r + (half << 3);
      float hb = xt[rr * VLDW + 128 + o0 + lm];
      float v = hb + fmaxf(acc[r] + bv, 0.f);
      long m = (mt << 4) + rr;
      out[(m << 7) + o0 + lm] = v;
      lsum += v; lsq += v * v;
    }
  }
  atomicAdd(&bnSum[o0 + lm], lsum);
  atomicAdd(&bnSq[o0 + lm], lsq);
}

__global__ void zero_stats(float* p) { p[threadIdx.x] = 0.f; }

__global__ void bn_finalize(const float* __restrict__ sum,
                            const float* __restrict__ sq,
                            const float* __restrict__ gamma,
                            const float* __restrict__ beta,
                            float* __restrict__ scale,
                            float* __restrict__ shift) {
  int c = threadIdx.x;
  const float invM = 1.f / 147456.f;  // B*N*N
  float m = sum[c] * invM;
  float var = sq[c] * invM - m * m;
  float s = gamma[c] * rsqrtf(var + 1e-5f);
  scale[c] = s;
  shift[c] = beta[c] - m * s;
}

__global__ void bn_apply(float* __restrict__ out, const float* __restrict__ scale,
                         const float* __restrict__ shift, int n4) {
  int stride = gridDim.x * blockDim.x;
  for (int i = blockIdx.x * blockDim.x + threadIdx.x; i < n4; i += stride) {
    float4 v = ((float4*)out)[i];
    int c = (i & 31) << 2;
    float4 sc = *(const float4*)(scale + c);
    float4 sh = *(const float4*)(shift + c);
    v.x = v.x * sc.x + sh.x; v.y = v.y * sc.y + sh.y;
    v.z = v.z * sc.z + sh.z; v.w = v.w * sc.w + sh.w;
    ((float4*)out)[i] = v;
  }
}

extern "C" void kernel_launch(void* const* d_in, const int* in_sizes, int n_in,
                              void* d_out, int out_size, void* d_ws, size_t ws_size,
                              hipStream_t stream) {
  const float* e      = (const float*)d_in[0];
  const int*   knbr   = (const int*)d_in[1];
  const float* W1_w   = (const float*)d_in[2];
  const float* W1_b   = (const float*)d_in[3];
  const float* W2_w   = (const float*)d_in[4];
  const float* W2_b   = (const float*)d_in[5];
  const float* W3_w   = (const float*)d_in[6];
  const float* W3_b   = (const float*)d_in[7];
  const float* We_w   = (const float*)d_in[8];
  const float* We_b   = (const float*)d_in[9];
  const float* Vin_w  = (const float*)d_in[10];
  const float* Vin_b  = (const float*)d_in[11];
  const float* V_w    = (const float*)d_in[12];
  const float* V_b    = (const float*)d_in[13];
  const float* bn1_g  = (const float*)d_in[14];
  const float* bn1_b  = (const float*)d_in[15];
  const float* bn2_g  = (const float*)d_in[16];
  const float* bn2_b  = (const float*)d_in[17];
  float* out = (float*)d_out;

  // workspace layout (floats)
  float* f      = (float*)d_ws;
  float* kkbuf  = f;                        // 18432*128
  float* vvbuf  = kkbuf + 2359296;          // 18432*128
  float* qbuf   = vvbuf + 2359296;          // 147456*128 (becomes att in place)
  float* h1buf  = qbuf + 18874368;          // 147456*128
  float* stats  = h1buf + 18874368;         // sums + scale/shift
  float* sum1 = stats,       * sq1 = stats + 128;
  float* sum2 = stats + 256, * sq2 = stats + 384;
  float* scale1 = stats + 512, * shift1 = stats + 640;
  float* scale2 = stats + 768, * shift2 = stats + 896;

  zero_stats<<<1, 512, 0, stream>>>(stats);

  // q = e @ W1^T + b1
  gemm128<<<2304, 256, 0, stream>>>(e, W1_w, W1_b, qbuf, 9216, nullptr,
                                    nullptr, nullptr, nullptr);
  // kk = gather(e) @ W2^T + b2 ; vv = gather(e) @ W3^T + b3
  gemm128<<<1152, 256, 0, stream>>>(e, W2_w, W2_b, kkbuf, 1152, knbr,
                                    nullptr, nullptr, nullptr);
  gemm128<<<1152, 256, 0, stream>>>(e, W3_w, W3_b, vvbuf, 1152, knbr,
                                    nullptr, nullptr, nullptr);
  // attention (att written over qbuf)
  attn<<<768, 256, 0, stream>>>(qbuf, kkbuf, vvbuf);
  // h1 = e + relu(att @ We^T + be), BN1 sums
  gemm128<<<2304, 256, 0, stream>>>(qbuf, We_w, We_b, h1buf, 9216, nullptr,
                                    e, sum1, sq1);
  bn_finalize<<<1, 128, 0, stream>>>(sum1, sq1, bn1_g, bn1_b, scale1, shift1);
  // gate: h2 -> d_out, BN2 sums
  gate_fused<<<1152, 256, 0, stream>>>(e, h1buf, scale1, shift1, Vin_w, Vin_b,
                                       V_w, V_b, out, 9216, sum2, sq2);
  bn_finalize<<<1, 128, 0, stream>>>(sum2, sq2, bn2_g, bn2_b, scale2, shift2);
  // final BN2 affine in place on d_out
  bn_apply<<<2048, 256, 0, stream>>>(out, scale2, shift2, 4718592);
}